// FactorNN_81114752352750
// MI455X (gfx1250) — compile-verified
//
#include <hip/hip_runtime.h>

typedef __attribute__((ext_vector_type(2))) float v2f;
typedef __attribute__((ext_vector_type(4))) float v4f;
typedef __attribute__((ext_vector_type(8))) float v8f;

#define EPSN 1e-5f

// ---------------------------------------------------------------------------
// WMMA 1x1 conv:  y[b,o,m] (or channel-last y[b,m,o]) = act( sum_c W[o,c]*x[b,c,m] + bias[o] )
// flags: bit0 = relu, bit1 = accumulate into y, bit2 = channel-last output
//
// Block = 256 threads = 8 waves, all sharing one 16-row O-tile (blockIdx.y).
// The 16xC weight tile is staged once into LDS; each wave computes a
// 16(O) x 64(M) strip: 4 independent accumulators, 4 V_WMMA_F32_16X16X4_F32
// per K-step sharing a single A fragment (4x arithmetic per A fetch, 4
// independent WMMA chains for latency hiding). C is a template constant so
// the K-loop fully unrolls.
//
// A (16x4): lanes 0-15 rows 0-15 K=0,1 ; lanes 16-31 rows 0-15 K=2,3.
// B (4x16): lanes 0-15 cols 0-15 K=0,1 ; lanes 16-31 cols 0-15 K=2,3.
// D (16x16 f32, 8 VGPRs): vgpr r = row r (lanes 0-15) / r+8 (lanes 16-31), col = lane&15.
// ---------------------------------------------------------------------------
template <int C>
__global__ __launch_bounds__(256) void k_conv_wmma(
    const float* __restrict__ x, const float* __restrict__ W,
    const float* __restrict__ bias, float* __restrict__ y,
    int O, int M, int flags)
{
    __shared__ float sW[16 * C];
    const int ot = blockIdx.y << 4;
    // W rows of this O-tile are contiguous: W[ot*C .. (ot+16)*C)
    for (int i = threadIdx.x; i < 16 * C; i += 256)
        sW[i] = W[(size_t)ot * C + i];
    __syncthreads();

    const int lane = threadIdx.x & 31;
    const int wave = threadIdx.x >> 5;
    const int mt0  = (blockIdx.x * 8 + wave) << 6;   // 64 columns per wave
    const int b    = blockIdx.z;
    if (mt0 >= M) return;                            // wave-uniform (grids divide exactly)

    const float* xb  = x + (size_t)b * C * M;
    const int half = lane >> 4;
    const int l16  = lane & 15;
    const int kh   = half << 1;

    v8f acc[4];
#pragma unroll
    for (int u = 0; u < 4; ++u) acc[u] = (v8f){0.f, 0.f, 0.f, 0.f, 0.f, 0.f, 0.f, 0.f};

#pragma unroll
    for (int c0 = 0; c0 < C; c0 += 4) {
        v2f a;
        a.x = sW[l16 * C + c0 + kh];
        a.y = sW[l16 * C + c0 + kh + 1];
        const float* xr = xb + (size_t)(c0 + kh) * M + (mt0 + l16);
        v2f b0, b1, b2, b3;
        b0.x = xr[0];  b0.y = xr[M];
        b1.x = xr[16]; b1.y = xr[M + 16];
        b2.x = xr[32]; b2.y = xr[M + 32];
        b3.x = xr[48]; b3.y = xr[M + 48];
        acc[0] = __builtin_amdgcn_wmma_f32_16x16x4_f32(false, a, false, b0, (short)0, acc[0], false, false);
        acc[1] = __builtin_amdgcn_wmma_f32_16x16x4_f32(false, a, false, b1, (short)0, acc[1], false, false);
        acc[2] = __builtin_amdgcn_wmma_f32_16x16x4_f32(false, a, false, b2, (short)0, acc[2], false, false);
        acc[3] = __builtin_amdgcn_wmma_f32_16x16x4_f32(false, a, false, b3, (short)0, acc[3], false, false);
    }

    const int  rbase  = half << 3;
    const bool relu   = flags & 1;
    const bool accum  = flags & 2;
    const bool chlast = flags & 4;
#pragma unroll
    for (int u = 0; u < 4; ++u) {
        const int m = mt0 + (u << 4) + l16;
#pragma unroll
        for (int r = 0; r < 8; ++r) {
            const int o = ot + rbase + r;
            float val = acc[u][r] + bias[o];
            if (relu) val = fmaxf(val, 0.f);
            const size_t oi = chlast ? ((size_t)b * M + m) * (size_t)O + o
                                     : ((size_t)b * O + o) * (size_t)M + m;
            if (accum) y[oi] += val;
            else       y[oi]  = val;
        }
    }
}

// ---------------------------------------------------------------------------
// Instance-norm statistics: one block per (b,c) row; stats[bc] = {mean, rstd}
// ---------------------------------------------------------------------------
__global__ void k_istats(const float* __restrict__ x, float* __restrict__ stats, int M)
{
    __shared__ float sh0[256], sh1[256];
    const int bc = blockIdx.x;
    const float* row = x + (size_t)bc * M;
    float s = 0.f, s2 = 0.f;
    for (int i = threadIdx.x; i < M; i += blockDim.x) {
        float v = row[i]; s += v; s2 += v * v;
    }
    sh0[threadIdx.x] = s; sh1[threadIdx.x] = s2;
    __syncthreads();
    for (int st = blockDim.x >> 1; st > 0; st >>= 1) {
        if ((int)threadIdx.x < st) {
            sh0[threadIdx.x] += sh0[threadIdx.x + st];
            sh1[threadIdx.x] += sh1[threadIdx.x + st];
        }
        __syncthreads();
    }
    if (threadIdx.x == 0) {
        const float inv = 1.f / (float)M;
        const float mu  = sh0[0] * inv;
        const float var = fmaxf(sh1[0] * inv - mu * mu, 0.f);
        stats[2 * bc]     = mu;
        stats[2 * bc + 1] = rsqrtf(var + EPSN);
    }
}

// y = relu((x - mu) * rstd)  [optionally accumulated into y]
__global__ void k_iapply(const float* __restrict__ x, const float* __restrict__ stats,
                         float* __restrict__ y, int M, int accum)
{
    const int bc = blockIdx.y;
    const size_t base = (size_t)bc * M;
    const float mu = stats[2 * bc], rs = stats[2 * bc + 1];
    const int i = blockIdx.x * blockDim.x + threadIdx.x;
    if (i >= M) return;
    const float v = fmaxf((x[base + i] - mu) * rs, 0.f);
    if (accum) y[base + i] += v;
    else       y[base + i]  = v;
}

// ---------------------------------------------------------------------------
// Batch-norm (training stats over b and m, per channel c)
// ---------------------------------------------------------------------------
__global__ void k_bstats(const float* __restrict__ x, float* __restrict__ stats,
                         int Bn, int C, int M)
{
    __shared__ float sh0[256], sh1[256];
    const int c = blockIdx.x;
    float s = 0.f, s2 = 0.f;
    for (int b = 0; b < Bn; ++b) {
        const float* row = x + ((size_t)b * C + c) * M;
        for (int i = threadIdx.x; i < M; i += blockDim.x) {
            float v = row[i]; s += v; s2 += v * v;
        }
    }
    sh0[threadIdx.x] = s; sh1[threadIdx.x] = s2;
    __syncthreads();
    for (int st = blockDim.x >> 1; st > 0; st >>= 1) {
        if ((int)threadIdx.x < st) {
            sh0[threadIdx.x] += sh0[threadIdx.x + st];
            sh1[threadIdx.x] += sh1[threadIdx.x + st];
        }
        __syncthreads();
    }
    if (threadIdx.x == 0) {
        const float inv = 1.f / (float)(Bn * M);
        const float mu  = sh0[0] * inv;
        const float var = fmaxf(sh1[0] * inv - mu * mu, 0.f);
        stats[2 * c]     = mu;
        stats[2 * c + 1] = rsqrtf(var + EPSN);
    }
}

__global__ void k_bapply(const float* __restrict__ x, const float* __restrict__ stats,
                         const float* __restrict__ gamma, const float* __restrict__ beta,
                         float* __restrict__ y, int C, int M)
{
    const int bc = blockIdx.y;
    const int c  = bc % C;
    const float mu = stats[2 * c], rs = stats[2 * c + 1];
    const size_t base = (size_t)bc * M;
    const int i = blockIdx.x * blockDim.x + threadIdx.x;
    if (i >= M) return;
    const float v = (x[base + i] - mu) * rs * gamma[c] + beta[c];
    y[base + i] = fmaxf(v, 0.f);
}

// ---------------------------------------------------------------------------
// Gather + etype-weighted sum over T + max over K.
// yt is channel-last [B][Msrc][T*O] so one neighbor row is T*O contiguous floats:
//   agg[b,o,n] = max_k sum_t et[b,t,n,k] * yt[b, idx[b,n,k], t*O + o]
// One wave per (b,n); lane owns O/32 consecutive channels (float4/float2 row reads).
// ---------------------------------------------------------------------------
template <int O>
__global__ void k_gathermax(const float* __restrict__ yt, const int* __restrict__ idx,
                            const float* __restrict__ et, float* __restrict__ agg,
                            int Msrc, int Ndst)
{
    constexpr int VL = O / 32;
    const int lane = threadIdx.x & 31;
    const int wave = threadIdx.x >> 5;
    const int n = blockIdx.x * (blockDim.x >> 5) + wave;
    const int b = blockIdx.y;
    if (n >= Ndst) return;

    const int*   ip = idx + ((size_t)b * Ndst + n) * 16;
    const float* ep = et + ((size_t)(b * 4) * Ndst + n) * 16;
    const size_t estr = (size_t)Ndst * 16;

    float acc[VL];
#pragma unroll
    for (int v = 0; v < VL; ++v) acc[v] = -3.402823466e+38f;

    int jn = ip[0];
    for (int k = 0; k < 16; ++k) {
        const int j = jn;
        if (k < 15) {
            jn = ip[k + 1];
            __builtin_prefetch(yt + ((size_t)b * Msrc + jn) * (size_t)(4 * O), 0, 0);
        }
        const float e0 = ep[k];
        const float e1 = ep[estr + k];
        const float e2 = ep[2 * estr + k];
        const float e3 = ep[3 * estr + k];
        const float* row = yt + ((size_t)b * Msrc + j) * (size_t)(4 * O);

        float s[VL];
#pragma unroll
        for (int v = 0; v < VL; ++v) s[v] = 0.f;
#pragma unroll
        for (int t = 0; t < 4; ++t) {
            const float e = (t == 0) ? e0 : (t == 1) ? e1 : (t == 2) ? e2 : e3;
            if constexpr (VL == 4) {
                v4f d = *(const v4f*)(row + t * O + lane * 4);
                s[0] += e * d.x; s[1] += e * d.y; s[2] += e * d.z; s[3] += e * d.w;
            } else {
                v2f d = *(const v2f*)(row + t * O + lane * 2);
                s[0] += e * d.x; s[1] += e * d.y;
            }
        }
#pragma unroll
        for (int v = 0; v < VL; ++v) acc[v] = fmaxf(acc[v], s[v]);
    }
#pragma unroll
    for (int v = 0; v < VL; ++v)
        agg[((size_t)b * O + (lane * VL + v)) * (size_t)Ndst + n] = acc[v];
}

// out[b,n] = sum_c W[c]*y[b,c,n] + bias  (O=1 head)
__global__ void k_fc2(const float* __restrict__ y, const float* __restrict__ W,
                      const float* __restrict__ bias, float* __restrict__ out,
                      int C, int M)
{
    const int i = blockIdx.x * blockDim.x + threadIdx.x;
    const int b = blockIdx.y;
    if (i >= M) return;
    float s = bias[0];
    for (int c = 0; c < C; ++c) s += W[c] * y[((size_t)b * C + c) * M + i];
    out[(size_t)b * M + i] = s;
}

__global__ void k_add(float* __restrict__ dst, const float* __restrict__ src, size_t n)
{
    const size_t i = (size_t)blockIdx.x * blockDim.x + threadIdx.x;
    if (i < n) dst[i] += src[i];
}

// ---------------------------------------------------------------------------
// Host orchestration
// ---------------------------------------------------------------------------
static inline void conv(hipStream_t s, const float* x, const float* W, const float* b,
                        float* y, int Bn, int C, int O, int M, int flags)
{
    dim3 grid(M / 512, O / 16, Bn);   // 8 waves/block, each wave: 16(O) x 64(M)
    switch (C) {
    case 8:   k_conv_wmma<8><<<grid, 256, 0, s>>>(x, W, b, y, O, M, flags); break;
    case 16:  k_conv_wmma<16><<<grid, 256, 0, s>>>(x, W, b, y, O, M, flags); break;
    case 64:  k_conv_wmma<64><<<grid, 256, 0, s>>>(x, W, b, y, O, M, flags); break;
    default:  k_conv_wmma<128><<<grid, 256, 0, s>>>(x, W, b, y, O, M, flags); break;
    }
}

static inline void inorm_relu(hipStream_t s, const float* x, float* y, float* stats,
                              int Bn, int C, int M, int accum)
{
    k_istats<<<dim3(Bn * C), 256, 0, s>>>(x, stats, M);
    k_iapply<<<dim3(M / 256, Bn * C), 256, 0, s>>>(x, stats, y, M, accum);
}

extern "C" void kernel_launch(void* const* d_in, const int* in_sizes, int n_in,
                              void* d_out, int out_size, void* d_ws, size_t ws_size,
                              hipStream_t stream)
{
    (void)in_sizes; (void)n_in; (void)out_size; (void)ws_size;
    const int B = 2, N = 16384, F = 8192;

    const float* node      = (const float*)d_in[0];
    const float* hop[2]    = {(const float*)d_in[1], (const float*)d_in[2]};
    const int*   idx_f2v[2] = {(const int*)d_in[3], (const int*)d_in[4]};
    const int*   idx_v2f[2] = {(const int*)d_in[5], (const int*)d_in[6]};
    const float* et_f2v[2] = {(const float*)d_in[7], (const float*)d_in[8]};
    const float* et_v2f[2] = {(const float*)d_in[9], (const float*)d_in[10]};

    // params leaves, jax pytree flatten order (dict keys sorted, list order):
    // factor_map, fc1, fc2, layers[0]{f2f,f2v,v2f,v2v}, layers[1]{f2f,f2v(res),v2f(res),v2v}, node_map
    int p = 11;
    auto nxt = [&]() { return (const float*)d_in[p++]; };

    const float *fm_W[2], *fm_b[2], *fm_beta[2], *fm_gamma[2];
    for (int j = 0; j < 2; ++j) { fm_W[j] = nxt(); fm_b[j] = nxt(); fm_beta[j] = nxt(); fm_gamma[j] = nxt(); }
    const float* fc1_W = nxt(); const float* fc1_b = nxt();
    const float* fc2_W = nxt(); const float* fc2_b = nxt();

    const float *l0_f2f_W[2], *l0_f2f_b[2];
    for (int j = 0; j < 2; ++j) { l0_f2f_W[j] = nxt(); l0_f2f_b[j] = nxt(); }
    const float *l0_f2v_Wt[2], *l0_f2v_bt[2];
    for (int j = 0; j < 2; ++j) { l0_f2v_Wt[j] = nxt(); l0_f2v_bt[j] = nxt(); }
    const float *l0_v2f_Wt[2], *l0_v2f_bt[2];
    for (int j = 0; j < 2; ++j) { l0_v2f_Wt[j] = nxt(); l0_v2f_bt[j] = nxt(); }
    const float* l0_v2v_W = nxt(); const float* l0_v2v_b = nxt();

    const float *l1_f2f_W[2], *l1_f2f_b[2];
    for (int j = 0; j < 2; ++j) { l1_f2f_W[j] = nxt(); l1_f2f_b[j] = nxt(); }
    struct Res { const float *mp_Wt, *mp_bt, *post_W, *post_b, *pre_W, *pre_b; };
    Res l1_f2v[2], l1_v2f[2];
    for (int j = 0; j < 2; ++j) l1_f2v[j] = Res{nxt(), nxt(), nxt(), nxt(), nxt(), nxt()};
    for (int j = 0; j < 2; ++j) l1_v2f[j] = Res{nxt(), nxt(), nxt(), nxt(), nxt(), nxt()};
    const float* l1_v2v_W = nxt(); const float* l1_v2v_b = nxt();
    const float* node_W = nxt(); const float* node_b = nxt();

    // -------- workspace layout (floats) --------
    float* ws = (float*)d_ws;
    const size_t VN = (size_t)B * 128 * N;    // 4,194,304
    const size_t VF = (size_t)B * 128 * F;    // 2,097,152
    float* bufA = ws;            // v
    float* bufB = bufA + VN;     // nv
    float* bufC = bufB + VN;     // N-side temp (agg / pre)
    float* bufD = bufC + VN;     // N-side temp (head)
    float* h0   = bufD + VN;
    float* h1   = h0 + VF;
    float* nh0  = h1 + VF;
    float* nh1  = nh0 + VF;
    float* bufF = nh1 + VF;      // F-side temp (agg)
    float* bufG = bufF + VF;     // F-side temp (pre)
    float* yt   = bufG + VF;     // [B][M][T*O] pre-transformed edge features (<= 4*VN)
    float* stats = yt + 4 * VN;  // 1024 floats of norm stats

    float* vcur = bufA; float* nv = bufB;
    float* hcur[2] = {h0, h1};  float* nh[2] = {nh0, nh1};

    // -------- embeddings --------
    conv(stream, node, node_W, node_b, vcur, B, 8, 64, N, /*relu*/1);
    for (int j = 0; j < 2; ++j) {
        conv(stream, hop[j], fm_W[j], fm_b[j], hcur[j], B, 16, 64, F, 0);
        k_bstats<<<dim3(64), 256, 0, stream>>>(hcur[j], stats, B, 64, F);
        k_bapply<<<dim3(F / 256, B * 64), 256, 0, stream>>>(hcur[j], stats,
                                                            fm_gamma[j], fm_beta[j],
                                                            hcur[j], 64, F);
    }

    // -------- layer 0: 64 -> 128 --------
    conv(stream, vcur, l0_v2v_W, l0_v2v_b, nv, B, 64, 128, N, 0);
    inorm_relu(stream, nv, nv, stats, B, 128, N, 0);
    for (int j = 0; j < 2; ++j) {
        conv(stream, hcur[j], l0_f2f_W[j], l0_f2f_b[j], nh[j], B, 64, 128, F, 0);
        inorm_relu(stream, nh[j], nh[j], stats, B, 128, F, 0);
    }
    for (int j = 0; j < 2; ++j) {
        // f2v: pre-transform all 4 edge types as one O=512 GEMM (channel-last), gather on N
        conv(stream, hcur[j], l0_f2v_Wt[j], l0_f2v_bt[j], yt, B, 64, 512, F, /*chlast*/4);
        k_gathermax<128><<<dim3(N / 8, B), 256, 0, stream>>>(yt, idx_f2v[j], et_f2v[j], bufC, F, N);
        inorm_relu(stream, bufC, nv, stats, B, 128, N, /*accum*/1);
        // v2f
        conv(stream, vcur, l0_v2f_Wt[j], l0_v2f_bt[j], yt, B, 64, 512, N, 4);
        k_gathermax<128><<<dim3(F / 8, B), 256, 0, stream>>>(yt, idx_v2f[j], et_v2f[j], bufF, N, F);
        inorm_relu(stream, bufF, nh[j], stats, B, 128, F, 1);
    }
    { float* t = vcur; vcur = nv; nv = t; }                       // nin != nout: v = nv
    for (int j = 0; j < 2; ++j) { float* t = hcur[j]; hcur[j] = nh[j]; nh[j] = t; }

    // -------- layer 1: 128 -> 128 (residual blocks) --------
    conv(stream, vcur, l1_v2v_W, l1_v2v_b, nv, B, 128, 128, N, 0);
    inorm_relu(stream, nv, nv, stats, B, 128, N, 0);
    for (int j = 0; j < 2; ++j) {
        conv(stream, hcur[j], l1_f2f_W[j], l1_f2f_b[j], nh[j], B, 128, 128, F, 0);
        inorm_relu(stream, nh[j], nh[j], stats, B, 128, F, 0);
    }
    for (int j = 0; j < 2; ++j) {
        // f2v residual block: pre(128->64) -> mp(64->64, T folded, O=256) -> post(64->128) += nv
        conv(stream, hcur[j], l1_f2v[j].pre_W, l1_f2v[j].pre_b, bufG, B, 128, 64, F, 0);
        inorm_relu(stream, bufG, bufG, stats, B, 64, F, 0);
        conv(stream, bufG, l1_f2v[j].mp_Wt, l1_f2v[j].mp_bt, yt, B, 64, 256, F, 4);
        k_gathermax<64><<<dim3(N / 8, B), 256, 0, stream>>>(yt, idx_f2v[j], et_f2v[j], bufC, F, N);
        inorm_relu(stream, bufC, bufC, stats, B, 64, N, 0);
        conv(stream, bufC, l1_f2v[j].post_W, l1_f2v[j].post_b, nv, B, 64, 128, N, /*accum*/2);
        // v2f residual block
        conv(stream, vcur, l1_v2f[j].pre_W, l1_v2f[j].pre_b, bufC, B, 128, 64, N, 0);
        inorm_relu(stream, bufC, bufC, stats, B, 64, N, 0);
        conv(stream, bufC, l1_v2f[j].mp_Wt, l1_v2f[j].mp_bt, yt, B, 64, 256, N, 4);
        k_gathermax<64><<<dim3(F / 8, B), 256, 0, stream>>>(yt, idx_v2f[j], et_v2f[j], bufF, N, F);
        inorm_relu(stream, bufF, bufF, stats, B, 64, F, 0);
        conv(stream, bufF, l1_v2f[j].post_W, l1_v2f[j].post_b, nh[j], B, 64, 128, F, 2);
    }
    // residual: v = v + nv ; h = h + nh
    k_add<<<dim3((unsigned)(VN / 256)), 256, 0, stream>>>(nv, vcur, VN);
    { float* t = vcur; vcur = nv; nv = t; }
    for (int j = 0; j < 2; ++j) {
        k_add<<<dim3((unsigned)(VF / 256)), 256, 0, stream>>>(nh[j], hcur[j], VF);
        float* t = hcur[j]; hcur[j] = nh[j]; nh[j] = t;
    }

    // -------- head --------
    conv(stream, vcur, fc1_W, fc1_b, bufD, B, 128, 128, N, 0);
    inorm_relu(stream, bufD, bufD, stats, B, 128, N, 0);
    k_fc2<<<dim3(N / 256, B), 256, 0, stream>>>(bufD, fc2_W, fc2_b, (float*)d_out, 128, N);
}